// TimeSequenceWarping_34213709480496
// MI455X (gfx1250) — compile-verified
//
#include <hip/hip_runtime.h>

// Time-sequence warp (dense_image_warp on a height-2 lifted image, last row).
// out[b,t,:] = lerp(x[b,idx,:], x[b,idx+1,:], alpha)
//   q     = t - offset[b,t]
//   idx   = clip(floor(q), 0, T-2)
//   alpha = clip(q - idx, 0, 1)
//
// MI455X: pure bandwidth problem (~256 MB traffic, ~11 us @ 23.3 TB/s).
// wave32 layout: one wave per (b,t) row; 32 lanes x float4 = 512B coalesced
// segments, 4 segments per 512-float row. T/C are compile-time constants so
// row->(b,t) is shift/mask (no integer-division expansion) and the channel
// loop fully unrolls into b128 loads with immediate offsets. Non-temporal
// b128 stores keep the 128MB x tensor resident in the 192MB L2, and
// global_prefetch_b8 warms the gather window ~2 frames ahead.

typedef float v4f __attribute__((ext_vector_type(4)));

constexpr int kT     = 8192;   // frames (power of two; fixed by reference)
constexpr int kTLog2 = 13;
constexpr int kC     = 512;    // channels (fixed by reference)

__global__ __launch_bounds__(256, 4)
void TimeSequenceWarping_warp1d_kernel(const float* __restrict__ x,
                                       const float* __restrict__ offset,
                                       float* __restrict__ out,
                                       int rows) {
    const int wave = threadIdx.x >> 5;                 // 8 waves per block
    const int lane = threadIdx.x & 31;
    const int row  = blockIdx.x * 8 + wave;            // flat b*T + t
    if (row >= rows) return;

    const int t = row & (kT - 1);
    const int b = row >> kTLog2;

    // Per-row interpolation parameters (uniform across the wave; the offset
    // load broadcasts from cache).
    const float o = offset[row];
    const float q = (float)t - o;
    float f = floorf(q);
    f = fminf(fmaxf(f, 0.0f), (float)(kT - 2));        // TF clamp: [0, T-2]
    const float a = fminf(fmaxf(q - f, 0.0f), 1.0f);   // TF clamp: [0, 1]
    const int idx = (int)f;

    const float* __restrict__ Lrow = x + (((size_t)b << kTLog2) + (size_t)idx) * kC;
    const float* __restrict__ Rrow = Lrow + kC;
    float* __restrict__ Orow = out + (size_t)row * kC;

    const v4f* __restrict__ L = (const v4f*)Lrow;
    const v4f* __restrict__ R = (const v4f*)Rrow;
    v4f* __restrict__ O = (v4f*)Orow;

    // Speculative prefetch of row idx+2 (2 frames ahead of the gather window).
    // 32 lanes x 64B = 2KB = exactly one channel row. gfx1250 global_prefetch_b8;
    // invalid addresses are silently dropped.
    __builtin_prefetch((const char*)Rrow + (size_t)kC * sizeof(float) + lane * 64, 0, 0);

    constexpr int kVec = kC >> 2;                      // 128 float4 per row
    #pragma unroll
    for (int c = lane; c < kVec; c += 32) {            // exactly 4 iterations
        v4f l = L[c];
        v4f r = R[c];
        v4f res = l + a * (r - l);                     // v_fma per component
        __builtin_nontemporal_store(res, &O[c]);       // NT store: spare L2 for x
    }
}

extern "C" void kernel_launch(void* const* d_in, const int* in_sizes, int n_in,
                              void* d_out, int out_size, void* d_ws, size_t ws_size,
                              hipStream_t stream) {
    const float* x      = (const float*)d_in[0];   // (B, T, C) fp32
    const float* offset = (const float*)d_in[1];   // (B, T, 1) fp32
    float* out          = (float*)d_out;           // (B, T, C) fp32

    const int rows = in_sizes[1];                  // B*T (= 65536)

    const int rowsPerBlock = 8;                    // 8 waves of 32 = 256 threads
    const int grid = (rows + rowsPerBlock - 1) / rowsPerBlock;

    TimeSequenceWarping_warp1d_kernel<<<grid, 256, 0, stream>>>(x, offset, out, rows);
}